// GLSTMSMNIST_50087908606428
// MI455X (gfx1250) — compile-verified
//
#include <hip/hip_runtime.h>
#include <hip/hip_bf16.h>

typedef float v2f __attribute__((ext_vector_type(2)));
typedef float v8f __attribute__((ext_vector_type(8)));

#ifndef __has_builtin
#define __has_builtin(x) 0
#endif

__device__ __forceinline__ float tanh_fast(float v) {
#if __has_builtin(__builtin_amdgcn_tanhf)
  return __builtin_amdgcn_tanhf(v);   // v_tanh_f32 on gfx1250
#else
  return tanhf(v);
#endif
}
__device__ __forceinline__ float sigmoid_fast(float v) {
  return fmaf(0.5f, tanh_fast(0.5f * v), 0.5f);
}

constexpr int B_ = 131072, T_ = 28, G_ = 28, K_ = 4;
constexpr int ROWS = B_ * G_;   // 3,670,016 independent LSTM streams
constexpr int GK   = G_ * K_;   // 112 features into the FC

// -----------------------------------------------------------------------------
// Kernel 1: grouped-LSTM scan. One wave32 owns 16 (b,g) streams.
// Per step: gates_T(16x16) = Wh_perm(16x4) x h_T(4x16) + x (outer) wxin
// via one v_wmma_f32_16x16x4_f32. Gate rows are permuted so that
//   m = 4*u_pair + {0:i,1:f,2:g,3:o}, original j = (m&3)*4 + (m>>2).
// D layout: lane l (r=l&15, half=l>>4) holds gates for units (2*half, 2*half+1)
// of stream r in d[0..3], d[4..7]. New h lands exactly in B-matrix layout.
// -----------------------------------------------------------------------------
__global__ __launch_bounds__(256) void glstm_scan(
    const float* __restrict__ x, const float* __restrict__ W_in,
    const float* __restrict__ W_x, const float* __restrict__ W_h,
    float* __restrict__ hbuf) {
  const int lane = threadIdx.x & 31;
  const int wave = blockIdx.x * (blockDim.x >> 5) + (threadIdx.x >> 5);
  const int r    = lane & 15;
  const int half = lane >> 4;
  const int row  = wave * 16 + r;          // ROWS divisible by 16: no tail
  const int bidx = row / G_;
  const int gidx = row - bidx * G_;
  const float* xptr = x + (size_t)bidx * (T_ * G_) + gidx;

  // A matrix (loop-invariant): lane supplies A[m=r, k=2*half+{0,1}] = W_h[perm(r)][k]
  const int jm = ((r & 3) << 2) + (r >> 2);
  v2f a;
  a.x = W_h[jm * K_ + 2 * half];
  a.y = W_h[jm * K_ + 2 * half + 1];

  // wxin_perm[m] = sum_k W_x[perm(m)][k] * W_in[k]; lane needs m = 8*half + v
  const float win0 = W_in[0], win1 = W_in[1], win2 = W_in[2], win3 = W_in[3];
  float wxin[8];
#pragma unroll
  for (int v = 0; v < 8; ++v) {
    const int m = 8 * half + v;
    const int j = ((m & 3) << 2) + (m >> 2);
    wxin[v] = W_x[j * K_ + 0] * win0 + W_x[j * K_ + 1] * win1 +
              W_x[j * K_ + 2] * win2 + W_x[j * K_ + 3] * win3;
  }

  // Preload this stream's 28 inputs (lane r and r+16 read the same addresses).
  float xs[T_];
#pragma unroll
  for (int t = 0; t < T_; ++t) xs[t] = xptr[t * G_];

  v2f hvec = {0.f, 0.f};   // B-matrix: lane r -> h[r,0],h[r,1]; lane r+16 -> h[r,2],h[r,3]
  float cA = 0.f, cB = 0.f;

#pragma unroll
  for (int t = 0; t < T_; ++t) {
    // C = x (outer) wxin : lane covers C[m=8*half+v, n=r], needs x of stream r
    v8f cmat;
#pragma unroll
    for (int v = 0; v < 8; ++v) cmat[v] = xs[t] * wxin[v];

    const v8f d = __builtin_amdgcn_wmma_f32_16x16x4_f32(
        false, a, false, hvec, (short)0, cmat, false, false);

    // unit 2*half : d[0..3] = i,f,g,o
    const float iA = sigmoid_fast(d[0]), fA = sigmoid_fast(d[1]);
    const float gA = tanh_fast(d[2]),    oA = sigmoid_fast(d[3]);
    cA = fmaf(fA, cA, iA * gA);
    const float hA = oA * tanh_fast(cA);
    // unit 2*half+1 : d[4..7]
    const float iB = sigmoid_fast(d[4]), fB = sigmoid_fast(d[5]);
    const float gB = tanh_fast(d[6]),    oB = sigmoid_fast(d[7]);
    cB = fmaf(fB, cB, iB * gB);
    const float hB = oB * tanh_fast(cB);

    hvec.x = hA;
    hvec.y = hB;
  }

  // h_flat[row][2*half + {0,1}] — wave writes one contiguous 256B span
  *(v2f*)(hbuf + (size_t)row * K_ + 2 * half) = hvec;
}

// -----------------------------------------------------------------------------
// Kernel 2: FC  out(B,10) = h(B,112) x W_fc^T(112,10), 28 chunks of K=4 WMMA.
// A: lane loads h[b0+r, 4c+2*half+{0,1}]; B: W_fc[r, 4c+2*half+{0,1}] with
// rows >= 10 arithmetically zeroed (EXEC stays all-ones for WMMA).
// -----------------------------------------------------------------------------
__global__ __launch_bounds__(256) void glstm_fc(
    const float* __restrict__ hbuf, const float* __restrict__ W_fc,
    float* __restrict__ out) {
  const int lane = threadIdx.x & 31;
  const int wave = blockIdx.x * (blockDim.x >> 5) + (threadIdx.x >> 5);
  const int r    = lane & 15;
  const int half = lane >> 4;
  const int b0   = wave * 16;              // B divisible by 16: no tail
  const int wrow = (r < 10) ? r : 0;
  const float wscale = (r < 10) ? 1.f : 0.f;

  v8f acc = {0.f, 0.f, 0.f, 0.f, 0.f, 0.f, 0.f, 0.f};
#pragma unroll
  for (int c = 0; c < GK / K_; ++c) {      // 28 chunks
    const int k0 = 4 * c + 2 * half;
    v2f av = *(const v2f*)(hbuf + (size_t)(b0 + r) * GK + k0);
    v2f bv = *(const v2f*)(W_fc + (size_t)wrow * GK + k0);
    bv.x *= wscale;
    bv.y *= wscale;
    acc = __builtin_amdgcn_wmma_f32_16x16x4_f32(
        false, av, false, bv, (short)0, acc, false, false);
  }

  if (r < 10) {                            // D: lane holds out[b0+8*half+v, r]
#pragma unroll
    for (int v = 0; v < 8; ++v)
      out[(size_t)(b0 + 8 * half + v) * 10 + r] = acc[v];
  }
}

// -----------------------------------------------------------------------------
extern "C" void kernel_launch(void* const* d_in, const int* in_sizes, int n_in,
                              void* d_out, int out_size, void* d_ws, size_t ws_size,
                              hipStream_t stream) {
  const float* x    = (const float*)d_in[0];  // (B, T*G)
  const float* W_in = (const float*)d_in[1];  // (K, 1)
  const float* W_x  = (const float*)d_in[2];  // (4K, K)
  const float* W_h  = (const float*)d_in[3];  // (4K, K)
  const float* W_fc = (const float*)d_in[4];  // (10, G*K)
  float* hbuf = (float*)d_ws;                 // ROWS*K floats = 58.7 MB
  float* outp = (float*)d_out;                // (B, 10)

  // 8 waves/block, 16 streams/wave
  const int scan_blocks = ROWS / (16 * 8);    // 28672
  const int fc_blocks   = B_   / (16 * 8);    // 1024
  glstm_scan<<<scan_blocks, 256, 0, stream>>>(x, W_in, W_x, W_h, hbuf);
  glstm_fc<<<fc_blocks, 256, 0, stream>>>(hbuf, W_fc, outp);
}